// AttentiveLISTA_5308579578243
// MI455X (gfx1250) — compile-verified
//
#include <hip/hip_runtime.h>
#include <math.h>

// Problem constants (AttentiveLISTA reference)
#define Wd    128
#define Hd    128
#define HW    16384          // 128*128
#define NIMG  32             // B*T
#define CC    64             // K
#define CHALF 32             // K/2
#define DD    16             // D
#define NPIX  (NIMG * HW)    // 524288
#define NTILE (NPIX / 16)    // 32768 16-pixel tiles, 8 per row

typedef float v2f __attribute__((ext_vector_type(2)));
typedef float v8f __attribute__((ext_vector_type(8)));

// ---------------------------------------------------------------------------
// CDNA5 fp32 WMMA: D(16x16) = A(16x4) * B(4x16) + C  (wave32, VOP3P)
// A 16x4: lanes 0-15 hold K={k0,k0+1}, lanes 16-31 K={k0+2,k0+3}
// B 4x16 / C 16x16 mirrored per ISA 7.12.2
// ---------------------------------------------------------------------------
__device__ __forceinline__ v8f wmma4(v2f a, v2f b, v8f c) {
  return __builtin_amdgcn_wmma_f32_16x16x4_f32(false, a, false, b, (short)0, c,
                                               false, false);
}

__device__ __forceinline__ float sigmoidf_(float v) {
  return 1.f / (1.f + __expf(-v));
}

__device__ __forceinline__ int clamp127(int v) {
  return v < 0 ? 0 : (v > 127 ? 127 : v);
}

// Tile decode shared by all WMMA kernels: one wave = 16 consecutive w pixels.
#define TILE_DECODE()                                                          \
  const int lane  = threadIdx.x & 31;                                          \
  const int wslot = threadIdx.x >> 5;                                          \
  const int hh2   = lane >> 4;                                                 \
  const int mm    = lane & 15;                                                 \
  const int tile  = blockIdx.x * 8 + wslot;                                    \
  const int img   = tile >> 10;                                                \
  const int rp    = tile & 1023;                                               \
  const int hp    = rp >> 3;                                                   \
  const int wp0   = (rp & 7) << 4;                                             \
  const size_t pixb = (size_t)hp * Wd + wp0;

// ---------------------------------------------------------------------------
// conv1 (WMMA implicit GEMM): x [32,16,128,128] -> lbuf [32,64,128,128]
// GEMM [16px x 144] x [144 x 64], K ordered (tap, d); tap loop fully
// unrolled so dy/dx/bounds/base are loop-invariant; clamped loads + select
// instead of exec-predicated loads. K split in 2 passes (LDS < 64 KB).
// ---------------------------------------------------------------------------
__global__ void __launch_bounds__(256)
k_conv1(const float* __restrict__ x, const float* __restrict__ w,
        const float* __restrict__ b, float* __restrict__ lbuf) {
  __shared__ float wl[80 * 65];          // largest pass: taps 0..4 -> 80 rows
  __shared__ float bl[CC];
  __shared__ float stg[8 * 16 * 65];     // per-wave C staging
  if (threadIdx.x < CC) bl[threadIdx.x] = b[threadIdx.x];

  TILE_DECODE();
  const float* xi = x + (size_t)img * DD * HW;

  __syncthreads();
  v8f acc[4];
#pragma unroll
  for (int t = 0; t < 4; ++t) {
    float bv = bl[16 * t + mm];
#pragma unroll
    for (int r = 0; r < 8; ++r) acc[t][r] = bv;
  }

#pragma unroll
  for (int pass = 0; pass < 2; ++pass) {
    const int tapbeg = pass ? 5 : 0;
    const int tapend = pass ? 9 : 5;
    const int kbeg = tapbeg * DD;
    const int kcnt = (tapend - tapbeg) * DD;
    __syncthreads();
    for (int i = threadIdx.x; i < kcnt * CC; i += 256) {
      int lr = i >> 6, c = i & 63;
      int kk = kbeg + lr, tap = kk >> 4, d = kk & 15;
      wl[lr * 65 + c] = w[(c * DD + d) * 9 + tap];
    }
    __syncthreads();

#pragma unroll
    for (int tap = tapbeg; tap < tapend; ++tap) {
      const int dy = tap / 3 - 1, dx = tap % 3 - 1;   // constants after unroll
      int hh = hp + dy, ww = wp0 + mm + dx;
      bool ok = ((unsigned)hh < 128u) && ((unsigned)ww < 128u);
      const float* xb = xi + clamp127(hh) * Wd + clamp127(ww);
      const int lrb = (tap - tapbeg) * DD;
#pragma unroll
      for (int dq = 0; dq < 4; ++dq) {
        int d0  = 4 * dq + 2 * hh2;
        int lr0 = lrb + d0;
        float v0 = xb[(size_t)d0 * HW];          // always-safe clamped load
        float v1 = xb[(size_t)(d0 + 1) * HW];
        v2f a;
        a.x = ok ? v0 : 0.f;                     // cndmask, no branch
        a.y = ok ? v1 : 0.f;
#pragma unroll
        for (int t = 0; t < 4; ++t) {
          v2f bb;
          bb.x = wl[lr0 * 65 + 16 * t + mm];
          bb.y = wl[(lr0 + 1) * 65 + 16 * t + mm];
          acc[t] = wmma4(a, bb, acc[t]);
        }
      }
    }
  }

  // stage C-layout -> coalesced NCHW store
  float* sl = stg + wslot * (16 * 65);
#pragma unroll
  for (int t = 0; t < 4; ++t)
#pragma unroll
    for (int r = 0; r < 8; ++r)
      sl[(r + 8 * hh2) * 65 + 16 * t + mm] = acc[t][r];
  float* lo = lbuf + (size_t)img * CC * HW + pixb;
#pragma unroll 8
  for (int cc = 0; cc < 32; ++cc) {
    int c = 2 * cc + hh2;
    lo[(size_t)c * HW + mm] = sl[mm * 65 + c];
  }
}

// ---------------------------------------------------------------------------
// res3 (WMMA implicit GEMM): h = conv3x3(relu(l)), 64 -> 32
// GEMM [16px x 576] x [576 x 32], K ordered (tap, ci); tap-outer loop with
// invariant bounds/base; clamped loads + select; 2 passes over taps.
// ---------------------------------------------------------------------------
__global__ void __launch_bounds__(256)
k_res3(const float* __restrict__ lbuf, const float* __restrict__ w,
       float* __restrict__ hbuf) {
  __shared__ float wl[320 * 33];         // largest pass: taps 0..4 -> 320 rows
  __shared__ float stg[8 * 16 * 33];

  TILE_DECODE();
  const float* li = lbuf + (size_t)img * CC * HW;

  v8f acc0 = {}, acc1 = {};
#pragma unroll
  for (int pass = 0; pass < 2; ++pass) {
    const int tapbeg = pass ? 5 : 0;
    const int tapend = pass ? 9 : 5;
    const int kbeg = tapbeg * CC;
    const int kcnt = (tapend - tapbeg) * CC;
    __syncthreads();
    for (int i = threadIdx.x; i < kcnt * CHALF; i += 256) {
      int lr = i >> 5, c = i & 31;
      int kk = kbeg + lr, tap = kk >> 6, ci = kk & 63;
      wl[lr * 33 + c] = w[(c * CC + ci) * 9 + tap];
    }
    __syncthreads();

#pragma unroll
    for (int tap = tapbeg; tap < tapend; ++tap) {
      const int dy = tap / 3 - 1, dx = tap % 3 - 1;   // constants after unroll
      int hh = hp + dy, ww = wp0 + mm + dx;
      bool ok = ((unsigned)hh < 128u) && ((unsigned)ww < 128u);
      const float* lb = li + clamp127(hh) * Wd + clamp127(ww);
      const int lrb = (tap - tapbeg) * CC;
#pragma unroll 4
      for (int cq = 0; cq < 16; ++cq) {
        int ci0 = 4 * cq + 2 * hh2;
        int lr0 = lrb + ci0;
        float v0 = lb[(size_t)ci0 * HW];
        float v1 = lb[(size_t)(ci0 + 1) * HW];
        v2f a;
        a.x = ok ? fmaxf(v0, 0.f) : 0.f;   // relu + pad select fused
        a.y = ok ? fmaxf(v1, 0.f) : 0.f;
        v2f b0, b1;
        b0.x = wl[lr0 * 33 + mm];        b0.y = wl[(lr0 + 1) * 33 + mm];
        b1.x = wl[lr0 * 33 + 16 + mm];   b1.y = wl[(lr0 + 1) * 33 + 16 + mm];
        acc0 = wmma4(a, b0, acc0);
        acc1 = wmma4(a, b1, acc1);
      }
    }
  }

  float* sl = stg + wslot * (16 * 33);
#pragma unroll
  for (int r = 0; r < 8; ++r) {
    sl[(r + 8 * hh2) * 33 + mm]      = acc0[r];
    sl[(r + 8 * hh2) * 33 + 16 + mm] = acc1[r];
  }
  float* ho = hbuf + (size_t)img * CHALF * HW + pixb;
#pragma unroll 4
  for (int cc = 0; cc < 16; ++cc) {
    int c = 2 * cc + hh2;
    ho[(size_t)c * HW + mm] = sl[mm * 33 + c];
  }
}

// ---------------------------------------------------------------------------
// res1x1 (WMMA GEMM): l += conv1x1(relu(h)), 32 -> 64
// GEMM [16px x 32] x [32 x 64]; residual folded into accumulator init.
// ---------------------------------------------------------------------------
__global__ void __launch_bounds__(256)
k_res1x1(const float* __restrict__ hbuf, const float* __restrict__ w,
         float* __restrict__ lbuf) {
  __shared__ float wl[CHALF * 65];
  __shared__ float stg[8 * 16 * 65];
  for (int i = threadIdx.x; i < CHALF * CC; i += 256) {
    int ci = i >> 6, c = i & 63;
    wl[ci * 65 + c] = w[c * CHALF + ci];
  }
  __syncthreads();

  TILE_DECODE();
  const float* li = lbuf + (size_t)img * CC * HW + pixb;
  const float* hi = hbuf + (size_t)img * CHALF * HW + pixb;

  // acc = l (residual) in C layout
  v8f acc[4];
#pragma unroll
  for (int t = 0; t < 4; ++t) {
    const float* lc = li + (size_t)(16 * t + mm) * HW + 8 * hh2;
#pragma unroll
    for (int r = 0; r < 8; ++r) acc[t][r] = lc[r];
  }

#pragma unroll
  for (int q = 0; q < 8; ++q) {
    int k0 = 4 * q + 2 * hh2;
    v2f a;
    a.x = fmaxf(hi[(size_t)k0 * HW + mm], 0.f);
    a.y = fmaxf(hi[(size_t)(k0 + 1) * HW + mm], 0.f);
#pragma unroll
    for (int t = 0; t < 4; ++t) {
      v2f bb;
      bb.x = wl[k0 * 65 + 16 * t + mm];
      bb.y = wl[(k0 + 1) * 65 + 16 * t + mm];
      acc[t] = wmma4(a, bb, acc[t]);
    }
  }

  float* sl = stg + wslot * (16 * 65);
#pragma unroll
  for (int t = 0; t < 4; ++t)
#pragma unroll
    for (int r = 0; r < 8; ++r)
      sl[(r + 8 * hh2) * 65 + 16 * t + mm] = acc[t][r];
  float* lo = lbuf + (size_t)img * CC * HW + pixb;
#pragma unroll 8
  for (int cc = 0; cc < 32; ++cc) {
    int c = 2 * cc + hh2;
    lo[(size_t)c * HW + mm] = sl[mm * 65 + c];
  }
}

// ---------------------------------------------------------------------------
// channel attention: per (img,c) mean & max over HW
// ---------------------------------------------------------------------------
__global__ void __launch_bounds__(256)
k_ca_reduce(const float* __restrict__ lbuf, float* __restrict__ cav,
            float* __restrict__ cmx) {
  __shared__ float ssum[256];
  __shared__ float smax[256];
  const float* p = lbuf + (size_t)blockIdx.x * HW;
  float s = 0.f, m = -3.4e38f;
  for (int i = threadIdx.x; i < HW; i += 256) {
    float v = p[i];
    s += v;
    m = fmaxf(m, v);
  }
  ssum[threadIdx.x] = s;
  smax[threadIdx.x] = m;
  __syncthreads();
  for (int st = 128; st > 0; st >>= 1) {
    if (threadIdx.x < st) {
      ssum[threadIdx.x] += ssum[threadIdx.x + st];
      smax[threadIdx.x] = fmaxf(smax[threadIdx.x], smax[threadIdx.x + st]);
    }
    __syncthreads();
  }
  if (threadIdx.x == 0) {
    cav[blockIdx.x] = ssum[0] * (1.f / HW);
    cmx[blockIdx.x] = smax[0];
  }
}

// ---------------------------------------------------------------------------
// channel-attention FC: scale = sigmoid(fc(avg)+fc(max)); w1 [4,64], w2 [64,4]
// ---------------------------------------------------------------------------
__global__ void k_ca_fc(const float* __restrict__ cav,
                        const float* __restrict__ cmx,
                        const float* __restrict__ w1,
                        const float* __restrict__ w2,
                        float* __restrict__ scale) {
  int img = blockIdx.x;
  int c   = threadIdx.x;   // 0..63
  const float* a = cav + img * CC;
  const float* m = cmx + img * CC;
  float hsum[4];
#pragma unroll
  for (int j = 0; j < 4; ++j) {
    float sa = 0.f, sm = 0.f;
    for (int k = 0; k < CC; ++k) {
      float wv = w1[j * CC + k];
      sa = fmaf(a[k], wv, sa);
      sm = fmaf(m[k], wv, sm);
    }
    hsum[j] = fmaxf(sa, 0.f) + fmaxf(sm, 0.f);
  }
  float o = 0.f;
#pragma unroll
  for (int j = 0; j < 4; ++j) o = fmaf(hsum[j], w2[c * 4 + j], o);
  scale[img * CC + c] = sigmoidf_(o);
}

// ---------------------------------------------------------------------------
// spatial stats on channel-scaled maps: avg/max over channels per pixel
// ---------------------------------------------------------------------------
__global__ void __launch_bounds__(256)
k_sa_stats(const float* __restrict__ lbuf, const float* __restrict__ scale,
           float* __restrict__ amap, float* __restrict__ mmap) {
  int pix = blockIdx.x * 256 + threadIdx.x;
  int img = pix >> 14;
  int off = pix & (HW - 1);
  const float* li = lbuf + (size_t)img * CC * HW + off;
  const float* sc = scale + img * CC;
  float s = 0.f, m = -3.4e38f;
  for (int c = 0; c < CC; ++c) {
    float v = li[(size_t)c * HW] * sc[c];
    s += v;
    m = fmaxf(m, v);
  }
  amap[pix] = s * (1.f / CC);
  mmap[pix] = m;
}

// ---------------------------------------------------------------------------
// spatial-attention 3x3 conv (2->1) + sigmoid; threshold in place:
// l <- l * ca_scale * s / L
// ---------------------------------------------------------------------------
__global__ void __launch_bounds__(256)
k_thresh(float* __restrict__ lbuf, const float* __restrict__ scale,
         const float* __restrict__ amap, const float* __restrict__ mmap,
         const float* __restrict__ saw, const float* __restrict__ Lp) {
  int pix = blockIdx.x * 256 + threadIdx.x;
  int img = pix >> 14;
  int hp  = (pix >> 7) & 127;
  int wp  = pix & 127;
  int off = pix & (HW - 1);

  float acc = 0.f;
#pragma unroll
  for (int t = 0; t < 9; ++t) {
    int dy = t / 3 - 1, dx = t % 3 - 1;
    int hh = hp + dy, ww = wp + dx;
    if (hh >= 0 && hh < Hd && ww >= 0 && ww < Wd) {
      int n = img * HW + hh * Wd + ww;
      acc = fmaf(amap[n], saw[t], acc);       // sa_w[0][0][t]
      acc = fmaf(mmap[n], saw[9 + t], acc);   // sa_w[0][1][t]
    }
  }
  float f = sigmoidf_(acc) / (*Lp);
  const float* sc = scale + img * CC;
  float* li = lbuf + (size_t)img * CC * HW + off;
#pragma unroll
  for (int c = 0; c < CC; ++c) li[(size_t)c * HW] *= sc[c] * f;
}

// ---------------------------------------------------------------------------
// S = I - Dict^T Dict / L  (symmetric => transpose is a no-op); Dict -> d_out
// ---------------------------------------------------------------------------
__global__ void k_S_dict(const float* __restrict__ Dict,
                         const float* __restrict__ Lp,
                         float* __restrict__ Sbuf,
                         float* __restrict__ dout_dict) {
  int idx = blockIdx.x * 256 + threadIdx.x;
  if (idx < CC * CC) {
    int i = idx >> 6, j = idx & 63;
    float s = 0.f;
#pragma unroll
    for (int d = 0; d < DD; ++d) s = fmaf(Dict[d * CC + i], Dict[d * CC + j], s);
    Sbuf[idx] = (i == j ? 1.f : 0.f) - s / (*Lp);
  } else if (idx < CC * CC + DD * CC) {
    int k = idx - CC * CC;
    dout_dict[k] = Dict[k];
  }
}

// ---------------------------------------------------------------------------
// Fused LISTA kernel (WMMA fp32 16x16x4):
//   per 16-pixel tile: y = xn@Dict (WMMA), z0 = soft(y, l),
//   16x: z = soft(z@S + y/L, l)  [64 WMMA/iter/tile],
//   x_recon = z@Dict^T (WMMA); z/x_recon written coalesced via LDS staging.
// ---------------------------------------------------------------------------
__global__ void __launch_bounds__(256)
k_lista(const float* __restrict__ x, const float* __restrict__ lbuf,
        const float* __restrict__ Sg, const float* __restrict__ Dict,
        const float* __restrict__ Lp, float* __restrict__ zout,
        float* __restrict__ xrout) {
  __shared__ float Sl[CC * 65];        // S, row-padded (conflict-free)
  __shared__ float Dl[DD * 65];        // Dict, row-padded
  __shared__ float zst[8 * 16 * 65];   // per-wave z staging [pixel][chan]

  for (int i = threadIdx.x; i < CC * CC; i += 256)
    Sl[(i >> 6) * 65 + (i & 63)] = Sg[i];
  for (int i = threadIdx.x; i < DD * CC; i += 256)
    Dl[(i >> 6) * 65 + (i & 63)] = Dict[i];
  __syncthreads();

  const float invL = 1.f / (*Lp);
  TILE_DECODE();

  const float* xi = x    + (size_t)img * DD * HW + pixb;
  const float* li = lbuf + (size_t)img * CC * HW + pixb;
  float* zl = zst + wslot * (16 * 65);

  // thresholds in C-layout: element (row = r+8*hh2, col = 16t+mm)
  v8f lth[4];
#pragma unroll
  for (int t = 0; t < 4; ++t) {
    const float* lc = li + (size_t)(16 * t + mm) * HW + 8 * hh2;
#pragma unroll
    for (int r = 0; r < 8; ++r) lth[t][r] = lc[r];
  }

  // y = xn @ Dict  (A rows = pixels, K = D channels of x, NCHW strided)
  v8f z[4];
#pragma unroll
  for (int t = 0; t < 4; ++t) { v8f zz = {}; z[t] = zz; }
#pragma unroll
  for (int q = 0; q < 4; ++q) {
    int k0 = 4 * q + 2 * hh2;
    v2f a;
    a.x = xi[(size_t)k0 * HW + mm];
    a.y = xi[(size_t)(k0 + 1) * HW + mm];
#pragma unroll
    for (int t = 0; t < 4; ++t) {
      v2f b;
      b.x = Dl[k0 * 65 + 16 * t + mm];
      b.y = Dl[(k0 + 1) * 65 + 16 * t + mm];
      z[t] = wmma4(a, b, z[t]);
    }
  }
  v8f yl[4];
#pragma unroll
  for (int t = 0; t < 4; ++t) {
    yl[t] = z[t] * invL;                 // y/L term for iterations
#pragma unroll
    for (int r = 0; r < 8; ++r) {        // z0 = soft(y, l)
      float v = z[t][r];
      float mag = fmaxf(fabsf(v) - lth[t][r], 0.f);
      z[t][r] = copysignf(mag, v);
    }
  }

  // 16 LISTA iterations: z = soft(z@S + y/L, l)
  for (int it = 0; it < 16; ++it) {
    // stage z: C-layout regs -> LDS [pixel][chan] (padded 65)
#pragma unroll
    for (int t = 0; t < 4; ++t)
#pragma unroll
      for (int r = 0; r < 8; ++r)
        zl[(r + 8 * hh2) * 65 + 16 * t + mm] = z[t][r];

    v8f acc0 = yl[0], acc1 = yl[1], acc2 = yl[2], acc3 = yl[3];
#pragma unroll 4
    for (int q = 0; q < 16; ++q) {
      int k0 = 4 * q + 2 * hh2;
      v2f a;
      a.x = zl[mm * 65 + k0];
      a.y = zl[mm * 65 + k0 + 1];
      v2f b0, b1, b2, b3;
      b0.x = Sl[k0 * 65 + mm];        b0.y = Sl[(k0 + 1) * 65 + mm];
      b1.x = Sl[k0 * 65 + 16 + mm];   b1.y = Sl[(k0 + 1) * 65 + 16 + mm];
      b2.x = Sl[k0 * 65 + 32 + mm];   b2.y = Sl[(k0 + 1) * 65 + 32 + mm];
      b3.x = Sl[k0 * 65 + 48 + mm];   b3.y = Sl[(k0 + 1) * 65 + 48 + mm];
      acc0 = wmma4(a, b0, acc0);
      acc1 = wmma4(a, b1, acc1);
      acc2 = wmma4(a, b2, acc2);
      acc3 = wmma4(a, b3, acc3);
    }
    v8f acc[4] = {acc0, acc1, acc2, acc3};
#pragma unroll
    for (int t = 0; t < 4; ++t)
#pragma unroll
      for (int r = 0; r < 8; ++r) {
        float v = acc[t][r];
        float mag = fmaxf(fabsf(v) - lth[t][r], 0.f);
        z[t][r] = copysignf(mag, v);
      }
  }

  // restage final z, write z out coalesced (NCHW)
#pragma unroll
  for (int t = 0; t < 4; ++t)
#pragma unroll
    for (int r = 0; r < 8; ++r)
      zl[(r + 8 * hh2) * 65 + 16 * t + mm] = z[t][r];

  float* zo = zout + (size_t)img * CC * HW + pixb;
#pragma unroll 8
  for (int cc = 0; cc < 32; ++cc) {
    int c = 2 * cc + hh2;
    zo[(size_t)c * HW + mm] = zl[mm * 65 + c];
  }

  // x_recon = z @ Dict^T : B[k=c][n=d] = Dict[d][c] (contiguous rows of Dl)
  v8f xr = {};
#pragma unroll 4
  for (int q = 0; q < 16; ++q) {
    int k0 = 4 * q + 2 * hh2;
    v2f a;
    a.x = zl[mm * 65 + k0];
    a.y = zl[mm * 65 + k0 + 1];
    v2f b;
    b.x = Dl[mm * 65 + k0];
    b.y = Dl[mm * 65 + k0 + 1];
    xr = wmma4(a, b, xr);
  }
  // stage x_recon [pixel][d] and write coalesced (NCHW)
#pragma unroll
  for (int r = 0; r < 8; ++r) zl[(r + 8 * hh2) * 65 + mm] = xr[r];
  float* xo = xrout + (size_t)img * DD * HW + pixb;
#pragma unroll
  for (int dd = 0; dd < 8; ++dd) {
    int d = 2 * dd + hh2;
    xo[(size_t)d * HW + mm] = zl[mm * 65 + d];
  }
}

// ---------------------------------------------------------------------------
// launch
// ---------------------------------------------------------------------------
extern "C" void kernel_launch(void* const* d_in, const int* in_sizes, int n_in,
                              void* d_out, int out_size, void* d_ws,
                              size_t ws_size, hipStream_t stream) {
  const float* x     = (const float*)d_in[0];
  const float* Dict  = (const float*)d_in[1];
  const float* Lp    = (const float*)d_in[2];
  const float* convw = (const float*)d_in[3];
  const float* convb = (const float*)d_in[4];
  const float* r1w1  = (const float*)d_in[5];
  const float* r1w2  = (const float*)d_in[6];
  const float* r2w1  = (const float*)d_in[7];
  const float* r2w2  = (const float*)d_in[8];
  const float* caw1  = (const float*)d_in[9];
  const float* caw2  = (const float*)d_in[10];
  const float* saw   = (const float*)d_in[11];

  float* out        = (float*)d_out;
  float* zout       = out;                                   // [32,64,128,128]
  float* xrout      = zout + (size_t)NIMG * CC * HW;         // [32,16,128,128]
  float* dout_dict  = xrout + (size_t)NIMG * DD * HW;        // [16,64]

  // workspace (~206 MB): lbuf | hbuf | amap | mmap | cav | cmx | scale | S
  char* wsb   = (char*)d_ws;
  float* lbuf = (float*)wsb;
  float* hbuf = (float*)(wsb + (size_t)NIMG * CC * HW * 4);
  float* amap = (float*)(wsb + (size_t)NIMG * CC * HW * 4 +
                               (size_t)NIMG * CHALF * HW * 4);
  float* mmap  = amap + NPIX;
  float* cav   = mmap + NPIX;
  float* cmx   = cav + NIMG * CC;
  float* scale = cmx + NIMG * CC;
  float* Sbuf  = scale + NIMG * CC;

  dim3 blk(256);
  const int gtile = NTILE / 8;   // 4096 blocks, 8 waves (tiles) each
  k_conv1 <<<gtile, blk, 0, stream>>>(x, convw, convb, lbuf);
  k_res3  <<<gtile, blk, 0, stream>>>(lbuf, r1w1, hbuf);
  k_res1x1<<<gtile, blk, 0, stream>>>(hbuf, r1w2, lbuf);
  k_res3  <<<gtile, blk, 0, stream>>>(lbuf, r2w1, hbuf);
  k_res1x1<<<gtile, blk, 0, stream>>>(hbuf, r2w2, lbuf);
  k_ca_reduce<<<NIMG * CC, blk, 0, stream>>>(lbuf, cav, cmx);
  k_ca_fc    <<<NIMG, dim3(64), 0, stream>>>(cav, cmx, caw1, caw2, scale);
  k_sa_stats <<<NPIX / 256, blk, 0, stream>>>(lbuf, scale, amap, mmap);
  k_thresh   <<<NPIX / 256, blk, 0, stream>>>(lbuf, scale, amap, mmap, saw, Lp);
  k_S_dict   <<<20, blk, 0, stream>>>(Dict, Lp, Sbuf, dout_dict);
  k_lista    <<<gtile, blk, 0, stream>>>(x, lbuf, Sbuf, Dict, Lp, zout, xrout);
}